// DepthConsistencyLoss_9225589752244
// MI455X (gfx1250) — compile-verified
//
#include <hip/hip_runtime.h>
#include <math.h>

// ---------------------------------------------------------------------------
// DepthConsistencyLoss for MI455X (gfx1250, wave32).
//
// Roofline: dominant cost = streaming 128MB of points/densities (~5.5us at
// 23.3 TB/s) -> scatter kernel uses 3x float4 (global_load_b128) per thread
// (4 points / 64B per thread, fully coalesced). Scatter accumulators (2MB)
// are L2-resident (192MB L2).
// SSIM 11x11 box pooling = banded matmuls  S * (X * S) / 121  with
// S[k][j] = (|k-j|<=5), executed on V_WMMA_F32_16X16X4_F32 (fp32 WMMA keeps
// reference precision; S is exact 0/1, generated in registers).
//
// Workspace layout (floats), B = batch (4), HW = 65536:
//   [0,      BHW)   depth accumulator -> finalized depth
//   [BHW,   2BHW)   weight-sum accumulator
//   [2BHW,  3BHW)   normalized pred
//   [3BHW,  4BHW)   normalized gt
//   [4BHW,  5BHW)   valid mask
//   [5BHW, 10BHW)   horizontal-pooled intermediates, 5 quantities
//   [10BHW, +16)    per-batch min/max (uint-ordered float bits)
//   [10BHW+16, +3)  scalar accumulators: l1sum, masksum, ssimsum
// Total ~10.5 MB.
// ---------------------------------------------------------------------------

typedef __attribute__((ext_vector_type(2))) float v2f;
typedef __attribute__((ext_vector_type(8))) float v8f;

#define IMG_W 256
#define HW 65536
#define MIN_D 0.1f
#define MAX_D 10.0f
#define SSIM_C1 1.0e-4f
#define SSIM_C2 9.0e-4f

__device__ __forceinline__ v8f wmma4(v2f a, v2f b, v8f c) {
  // V_WMMA_F32_16X16X4_F32: D = A(16x4,f32) * B(4x16,f32) + C(16x16,f32)
  return __builtin_amdgcn_wmma_f32_16x16x4_f32(
      /*neg_a=*/false, a, /*neg_b=*/false, b,
      /*c_mod=*/(short)0, c, /*reuse_a=*/false, /*reuse_b=*/false);
}

__device__ __forceinline__ float band(int k, int j) {
  int d = k - j;
  return (d >= -5 && d <= 5) ? 1.0f : 0.0f;
}

__device__ __forceinline__ float wave_sum(float v) {
#pragma unroll
  for (int o = 16; o > 0; o >>= 1) v += __shfl_down(v, o, 32);
  return v;
}

// 8-byte aligned pair load (kk is always even -> o0 is even).
__device__ __forceinline__ v2f loadv2(const float* p, int o0) {
  return *reinterpret_cast<const v2f*>(p + o0);
}

// --- K0: zero accumulators, init min/max, zero scalars ----------------------
__global__ void k_init(float* acc2, unsigned* mm, float* sc, int zcount, int B) {
  int i = blockIdx.x * blockDim.x + threadIdx.x;
  if (i < zcount) acc2[i] = 0.0f;                            // depth + wsum
  if (i < 4 * B) mm[i] = ((i & 1) == 0) ? 0x7F800000u : 0u;  // min=+inf, max=0
  if (i < 3) sc[i] = 0.0f;
}

// --- K1: project + weighted scatter (bandwidth-dominant stage) --------------
// One thread per 4 points: three b128 point loads + one b128 density load.
__device__ __forceinline__ void scatter_one(float x, float y, float z, float d,
                                            int b, float* depth, float* wsum) {
  float w = 1.0f / (1.0f + expf(-d));  // sigmoid
  float zs = fmaxf(z, MIN_D);
  float u = (x / zs + 0.5f) * 256.0f;
  float v = (y / zs + 0.5f) * 256.0f;
  bool valid = (z > MIN_D) && (u >= 0.0f) && (u < 256.0f) &&
               (v >= 0.0f) && (v < 256.0f);
  if (valid) {
    int ui = min(max((int)floorf(u), 0), IMG_W - 1);
    int vi = min(max((int)floorf(v), 0), IMG_W - 1);
    int idx = b * HW + vi * IMG_W + ui;
    atomicAdd(&depth[idx], zs * w);
    atomicAdd(&wsum[idx], w);
  }
}

__global__ void k_scatter(const float4* __restrict__ pts4,
                          const float4* __restrict__ dens4,
                          float* __restrict__ depth, float* __restrict__ wsum,
                          long long total4, int N) {
  long long j = (long long)blockIdx.x * blockDim.x + threadIdx.x;
  if (j >= total4) return;
  __builtin_prefetch(pts4 + 3 * (j + 2048), 0, 1);  // global_prefetch_b8
  float4 p0 = pts4[3 * j + 0];
  float4 p1 = pts4[3 * j + 1];
  float4 p2 = pts4[3 * j + 2];
  float4 d4 = dens4[j];
  // N divisible by 4 -> 4 consecutive points share one batch index.
  int b = (int)((4 * j) / N);
  scatter_one(p0.x, p0.y, p0.z, d4.x, b, depth, wsum);
  scatter_one(p0.w, p1.x, p1.y, d4.y, b, depth, wsum);
  scatter_one(p1.z, p1.w, p2.x, d4.z, b, depth, wsum);
  scatter_one(p2.y, p2.z, p2.w, d4.w, b, depth, wsum);
}

// --- K2: depth = depth/wsum; per-batch valid min/max (uint-ordered) ---------
__global__ void k_finalize(float* __restrict__ depth,
                           const float* __restrict__ wsum,
                           const float* __restrict__ gt,
                           unsigned* __restrict__ mm, int BHW) {
  int i = blockIdx.x * blockDim.x + threadIdx.x;
  if (i >= BHW) return;
  int b = i >> 16;  // / HW
  unsigned* m = mm + b * 4;
  float ws = wsum[i];
  float d = (ws > 0.0f) ? depth[i] / ws : 0.0f;
  depth[i] = d;
  if (d > 0.0f) {  // positive floats: bit order == value order
    unsigned bits = __float_as_uint(d);
    atomicMin(&m[0], bits);
    atomicMax(&m[1], bits);
  }
  float g = gt[i];
  if (g > 0.0f) {
    unsigned bits = __float_as_uint(g);
    atomicMin(&m[2], bits);
    atomicMax(&m[3], bits);
  }
}

// --- K3: normalize, build mask, accumulate L1 and mask sums -----------------
__global__ void k_normalize(const float* __restrict__ depth,
                            const float* __restrict__ gt,
                            const unsigned* __restrict__ mm,
                            float* __restrict__ np, float* __restrict__ ng,
                            float* __restrict__ mask, float* __restrict__ sc,
                            int BHW) {
  int i = blockIdx.x * blockDim.x + threadIdx.x;
  if (i >= BHW) return;  // BHW multiple of 256: no partial waves
  int b = i >> 16;
  const unsigned* m = mm + b * 4;

  bool phas = (m[0] != 0x7F800000u);
  float pmin = phas ? fmaxf(__uint_as_float(m[0]), MIN_D) : 0.0f;
  float pmax = phas ? fminf(__uint_as_float(m[1]), MAX_D) : MAX_D;
  bool ghas = (m[2] != 0x7F800000u);
  float gmin = ghas ? fmaxf(__uint_as_float(m[2]), MIN_D) : 0.0f;
  float gmax = ghas ? fminf(__uint_as_float(m[3]), MAX_D) : MAX_D;

  float d = depth[i];
  float g = gt[i];
  float npv = (d - pmin) / (pmax - pmin + 1e-8f) * ((d > 0.0f) ? 1.0f : 0.0f);
  float ngv = (g - gmin) / (gmax - gmin + 1e-8f) * ((g > 0.0f) ? 1.0f : 0.0f);
  np[i] = npv;
  ng[i] = ngv;
  float mk = (npv > 0.0f && ngv > 0.0f) ? 1.0f : 0.0f;
  mask[i] = mk;

  float l1p = wave_sum(mk * fabsf(npv - ngv));
  float mks = wave_sum(mk);
  if ((threadIdx.x & 31) == 0) {
    atomicAdd(&sc[0], l1p);
    atomicAdd(&sc[1], mks);
  }
}

// --- K4: horizontal box pool via WMMA:  Hq = Xq * S -------------------------
// Templated on quantity Q (0:p 1:g 2:p*p 3:g*g 4:p*g) so the inner K-loop is
// branch-free. One wave per 16x16 output tile per batch.
// A = X rows (16x4 f32), B = band matrix S (4x16, analytic 0/1).
// A layout (ISA): lanes 0-15 K={0,1}, lanes 16-31 K={2,3}; B mirrors the
// half-wave K split; C/D: VGPR j -> M = j (+8 for upper half).
template <int Q>
__global__ void k_hpool(const float* __restrict__ np,
                        const float* __restrict__ ng,
                        float* __restrict__ hbase, int B) {
  int lane = threadIdx.x;
  int bid = blockIdx.x;
  int tile = bid & 255;
  int b = bid >> 8;
  int tr = tile >> 4, tc = tile & 15;
  int hp = lane >> 4, m = lane & 15;
  int row = tr * 16 + m;
  int colg = tc * 16 + m;  // N index for this lane (B operand & C/D column)

  const float* P = np + (size_t)b * HW;
  const float* G = ng + (size_t)b * HW;

  v8f acc = {0.f, 0.f, 0.f, 0.f, 0.f, 0.f, 0.f, 0.f};
  for (int kt = tc - 1; kt <= tc + 1; ++kt) {  // wave-uniform branch
    if (kt < 0 || kt > 15) continue;           // zero-padding at edges
#pragma unroll
    for (int s = 0; s < 4; ++s) {
      int kk = kt * 16 + s * 4 + hp * 2;
      int o0 = row * IMG_W + kk;
      v2f av;
      if (Q == 0)      { av = loadv2(P, o0); }
      else if (Q == 1) { av = loadv2(G, o0); }
      else if (Q == 2) { v2f p = loadv2(P, o0); av = p * p; }
      else if (Q == 3) { v2f g = loadv2(G, o0); av = g * g; }
      else             { v2f p = loadv2(P, o0); v2f g = loadv2(G, o0); av = p * g; }
      v2f bv = {band(kk, colg), band(kk + 1, colg)};
      acc = wmma4(av, bv, acc);
    }
  }
  float* Hq = hbase + ((size_t)Q * B + b) * HW;
#pragma unroll
  for (int j = 0; j < 8; ++j) {
    int r = tr * 16 + j + hp * 8;
    Hq[r * IMG_W + colg] = acc[j];
  }
}

// --- K5: vertical box pool via WMMA (Z = S * Hq) + SSIM map + reduce --------
__global__ void k_vpool_ssim(const float* __restrict__ hbase,
                             const float* __restrict__ mask,
                             float* __restrict__ ssimsum, int B) {
  int lane = threadIdx.x;
  int bid = blockIdx.x;
  int tile = bid & 255;
  int b = bid >> 8;
  int tr = tile >> 4, tc = tile & 15;
  int hp = lane >> 4, m = lane & 15;
  int col = tc * 16 + m;
  int irow = tr * 16 + m;  // A-matrix M index for this lane

  const float* H0 = hbase + ((size_t)0 * B + b) * HW;
  const float* H1 = hbase + ((size_t)1 * B + b) * HW;
  const float* H2 = hbase + ((size_t)2 * B + b) * HW;
  const float* H3 = hbase + ((size_t)3 * B + b) * HW;
  const float* H4 = hbase + ((size_t)4 * B + b) * HW;

  v8f acc0 = {0.f,0.f,0.f,0.f,0.f,0.f,0.f,0.f};
  v8f acc1 = acc0, acc2 = acc0, acc3 = acc0, acc4 = acc0;

  for (int kt = tr - 1; kt <= tr + 1; ++kt) {  // wave-uniform
    if (kt < 0 || kt > 15) continue;
#pragma unroll
    for (int s = 0; s < 4; ++s) {
      int kk = kt * 16 + s * 4 + hp * 2;
      v2f av = {band(irow, kk), band(irow, kk + 1)};
      int o0 = kk * IMG_W + col;
      int o1 = o0 + IMG_W;
      v2f b0 = {H0[o0], H0[o1]};  acc0 = wmma4(av, b0, acc0);
      v2f b1 = {H1[o0], H1[o1]};  acc1 = wmma4(av, b1, acc1);
      v2f b2 = {H2[o0], H2[o1]};  acc2 = wmma4(av, b2, acc2);
      v2f b3 = {H3[o0], H3[o1]};  acc3 = wmma4(av, b3, acc3);
      v2f b4 = {H4[o0], H4[o1]};  acc4 = wmma4(av, b4, acc4);
    }
  }

  const float inv121 = 1.0f / 121.0f;
  float part = 0.0f;
#pragma unroll
  for (int j = 0; j < 8; ++j) {
    int r = tr * 16 + j + hp * 8;
    float mu1 = acc0[j] * inv121;
    float mu2 = acc1[j] * inv121;
    float s1 = acc2[j] * inv121 - mu1 * mu1;
    float s2 = acc3[j] * inv121 - mu2 * mu2;
    float s12 = acc4[j] * inv121 - mu1 * mu2;
    float num = (2.0f * mu1 * mu2 + SSIM_C1) * (2.0f * s12 + SSIM_C2);
    float den = (mu1 * mu1 + mu2 * mu2 + SSIM_C1) * (s1 + s2 + SSIM_C2);
    float mv = mask[(size_t)b * HW + r * IMG_W + col];
    part += (num / den) * mv;
  }
  part = wave_sum(part);
  if (lane == 0) atomicAdd(ssimsum, part);
}

// --- K6: combine scalars ----------------------------------------------------
__global__ void k_final(const float* __restrict__ sc, float* __restrict__ out) {
  float msum = sc[1];
  float l1 = sc[0] / (msum + 1e-8f);
  float ssim_loss = 1.0f - sc[2] / (msum + 1e-8f);
  float total = fminf(0.8f * l1 + 0.2f * ssim_loss, 1.0f);
  out[0] = (msum < 10.0f) ? 0.0f : total;
}

extern "C" void kernel_launch(void* const* d_in, const int* in_sizes, int n_in,
                              void* d_out, int out_size, void* d_ws, size_t ws_size,
                              hipStream_t stream) {
  const float4* pts4 = (const float4*)d_in[0];   // [B, N, 3] viewed as float4
  const float4* dens4 = (const float4*)d_in[1];  // [B, N, 1] viewed as float4
  const float* gt = (const float*)d_in[2];       // [B, 1, 256, 256]
  float* out = (float*)d_out;

  int B = in_sizes[2] / HW;                      // 4
  long long total = (long long)in_sizes[1];      // B * N
  int N = (int)(total / B);
  long long total4 = total / 4;                  // B*N divisible by 4
  int BHW = B * HW;

  float* wsf = (float*)d_ws;
  float* depth = wsf;                            // + wsum contiguous after it
  float* wsum = wsf + (size_t)BHW;
  float* np = wsf + 2 * (size_t)BHW;
  float* ng = wsf + 3 * (size_t)BHW;
  float* mask = wsf + 4 * (size_t)BHW;
  float* hbase = wsf + 5 * (size_t)BHW;          // 5 * BHW floats
  unsigned* mm = (unsigned*)(wsf + 10 * (size_t)BHW);
  float* sc = wsf + 10 * (size_t)BHW + 16;

  int zcount = 2 * BHW;
  k_init<<<(zcount + 255) / 256, 256, 0, stream>>>(depth, mm, sc, zcount, B);
  k_scatter<<<(int)((total4 + 255) / 256), 256, 0, stream>>>(pts4, dens4,
                                                             depth, wsum,
                                                             total4, N);
  k_finalize<<<(BHW + 255) / 256, 256, 0, stream>>>(depth, wsum, gt, mm, BHW);
  k_normalize<<<(BHW + 255) / 256, 256, 0, stream>>>(depth, gt, mm, np, ng,
                                                     mask, sc, BHW);
  k_hpool<0><<<B * 256, 32, 0, stream>>>(np, ng, hbase, B);
  k_hpool<1><<<B * 256, 32, 0, stream>>>(np, ng, hbase, B);
  k_hpool<2><<<B * 256, 32, 0, stream>>>(np, ng, hbase, B);
  k_hpool<3><<<B * 256, 32, 0, stream>>>(np, ng, hbase, B);
  k_hpool<4><<<B * 256, 32, 0, stream>>>(np, ng, hbase, B);
  k_vpool_ssim<<<B * 256, 32, 0, stream>>>(hbase, mask, sc + 2, B);
  k_final<<<1, 1, 0, stream>>>(sc, out);
}